// SparseEncoder_45981919871119
// MI455X (gfx1250) — compile-verified
//
#include <hip/hip_runtime.h>
#include <hip/hip_bf16.h>

typedef __attribute__((ext_vector_type(2))) float v2f;
typedef __attribute__((ext_vector_type(8))) float v8f;

#define EPI_BN_RELU_MASK     0
#define EPI_BN_MASK_ADD_RELU 1

// ---------------------------------------------------------------------------
// Implicit-GEMM 3x3x3 conv on V_WMMA_F32_16X16X4_F32 (fp32-exact).
// One wave computes a 16-voxel x (16*NT)-outchannel tile.
// A: 16 voxels x 4 k  (NDHWC gather; OOB handled by clamped address + 0/1
//     multiply -- no EXEC divergence in the hot loop)
// B: 4 k x 16 cout    (weights pre-packed into fragment order: each 4x16
//     block is 64 contiguous floats indexed by lane*2+j -> one b64 load)
// ---------------------------------------------------------------------------
template <int CIN, int COUT, int NT>
__global__ __launch_bounds__(32) void conv3d_wmma(
    const float* __restrict__ in, float* __restrict__ out,
    const float* __restrict__ wp,   // packed weights
    const float* __restrict__ bn_s, const float* __restrict__ bn_b,
    const float* __restrict__ mask, const float* __restrict__ res,
    int ID, int IH, int IW, int OD, int OH, int OW,
    int SD, int SH, int SW, int PD, int PH, int PW, int mode) {
  const int lane = threadIdx.x;
  const int half = lane >> 4;   // K-pair selector for A/B fragments
  const int lm   = lane & 15;
  const int koff = 2 * half;
  const int KC   = CIN / 4;     // k-chunks per tap
  const int NTT  = COUT / 16;   // total 16-wide cout tiles

  const int nvox   = OD * OH * OW;
  const int vbase  = blockIdx.x * 16;
  const int cobase = blockIdx.y * (16 * NT);
  const int tile0  = cobase >> 4;

  // Per-lane A-row voxel (both lane halves address the same 16 voxels).
  const int mv = vbase + lm;
  const bool vok = (mv < nvox);
  const int v0 = vok ? mv : 0;
  const int od = v0 / (OH * OW);
  const int rr = v0 % (OH * OW);
  const int oh = rr / OW;
  const int ow = rr % OW;

  v8f acc[NT];
#pragma unroll
  for (int nt = 0; nt < NT; ++nt) acc[nt] = (v8f){};

  for (int kd = 0; kd < 3; ++kd) {
    for (int kh = 0; kh < 3; ++kh) {
      for (int kw = 0; kw < 3; ++kw) {
        const int id = od * SD - PD + kd;
        const int ih = oh * SH - PH + kh;
        const int iw = ow * SW - PW + kw;
        const bool ok = vok && (unsigned)id < (unsigned)ID &&
                        (unsigned)ih < (unsigned)IH && (unsigned)iw < (unsigned)IW;
        const float okf = ok ? 1.f : 0.f;                       // v_cndmask
        const int rowbase = ok ? ((id * IH + ih) * IW + iw) * CIN : 0;
        const int tap = (kd * 3 + kh) * 3 + kw;
        const float* wtap =
            wp + ((size_t)tap * KC * NTT + tile0) * 64 + lane * 2;
#pragma unroll
        for (int kc = 0; kc < KC; ++kc) {
          // Unconditional, clamped A load; zeroed by okf (no EXEC games).
          v2f a = *(const v2f*)(in + rowbase + kc * 4 + koff);
          a = a * okf;
          const float* wblk = wtap + (size_t)kc * NTT * 64;
#pragma unroll
          for (int nt = 0; nt < NT; ++nt) {
            v2f b = *(const v2f*)(wblk + nt * 64);              // one b64 load
            acc[nt] = __builtin_amdgcn_wmma_f32_16x16x4_f32(
                false, a, false, b, (short)0, acc[nt], false, false);
          }
        }
      }
    }
  }

  // Epilogue: lane owns column co; VGPR r holds voxel vbase + 8*half + r.
#pragma unroll
  for (int nt = 0; nt < NT; ++nt) {
    const int co = cobase + nt * 16 + lm;
    const float s  = bn_s[co];
    const float bb = bn_b[co];
#pragma unroll
    for (int r = 0; r < 8; ++r) {
      const int v = vbase + half * 8 + r;
      if (v < nvox) {
        float y = acc[nt][r] * s + bb;
        const float m = mask[v];
        if (mode == EPI_BN_RELU_MASK) {
          y = fmaxf(y, 0.f) * m;                                  // relu*mask
        } else {
          y = fmaxf(y * m + res[(size_t)v * COUT + co], 0.f);     // +x, relu
        }
        out[(size_t)v * COUT + co] = y;
      }
    }
  }
}

__global__ void zero_f32(float* __restrict__ p, int n) {
  int i = blockIdx.x * blockDim.x + threadIdx.x;
  if (i < n) p[i] = 0.f;
}

// Scatter points into the dense grid; Cin padded 3 -> 4 (channel 3 stays 0).
__global__ void scatter_pts(const float* __restrict__ feat,
                            const int* __restrict__ coors,
                            float* __restrict__ dense4,
                            float* __restrict__ mask, int npts, int D, int H,
                            int W) {
  int i = blockIdx.x * blockDim.x + threadIdx.x;
  if (i >= npts) return;
  const int b = coors[i * 4 + 0];
  const int z = coors[i * 4 + 1];
  const int y = coors[i * 4 + 2];
  const int x = coors[i * 4 + 3];
  const size_t idx = (((size_t)b * D + z) * H + y) * W + x;
  dense4[idx * 4 + 0] = feat[i * 3 + 0];
  dense4[idx * 4 + 1] = feat[i * 3 + 1];
  dense4[idx * 4 + 2] = feat[i * 3 + 2];
  mask[idx] = 1.f;
}

// Repack w_in (3,3,3,3,16) -> (27,4,16) with a zero ci=3 plane.
__global__ void pad_win(const float* __restrict__ w3, float* __restrict__ w4) {
  int i = blockIdx.x * blockDim.x + threadIdx.x;
  if (i >= 27 * 4 * 16) return;
  const int co = i % 16;
  const int t = i / 16;
  const int ci = t % 4;
  const int tap = t / 4;
  w4[i] = (ci < 3) ? w3[(tap * 3 + ci) * 16 + co] : 0.f;
}

// Pack (27*cin, cout) row-major weights into WMMA B-fragment order:
// wp[((tap*(cin/4)+kc)*(cout/16)+tile)*64 + lane*2 + j]
//   = w[(tap*cin + kc*4 + 2*(lane>>4) + j)*cout + tile*16 + (lane&15)]
__global__ void pack_wgt(const float* __restrict__ w, float* __restrict__ wp,
                         int cin, int cout) {
  int i = blockIdx.x * blockDim.x + threadIdx.x;
  const int total = 27 * cin * cout;
  if (i >= total) return;
  const int j = i & 1;
  const int lane = (i >> 1) & 31;
  const int bi = i >> 6;
  const int ntt = cout / 16;
  const int tile = bi % ntt;
  int t = bi / ntt;
  const int kc = t % (cin / 4);
  const int tap = t / (cin / 4);
  const int k = kc * 4 + 2 * (lane >> 4) + j;
  const int co = tile * 16 + (lane & 15);
  wp[i] = w[(tap * cin + k) * cout + co];
}

// mask_out = (sum over 3x3x3 window of mask_in) > 0
__global__ void mask_down(const float* __restrict__ mi, float* __restrict__ mo,
                          int ID, int IH, int IW, int OD, int OH, int OW,
                          int SD, int SH, int SW, int PD, int PH, int PW) {
  int v = blockIdx.x * blockDim.x + threadIdx.x;
  const int n = OD * OH * OW;
  if (v >= n) return;
  const int od = v / (OH * OW);
  int r = v % (OH * OW);
  const int oh = r / OW;
  const int ow = r % OW;
  float acc = 0.f;
  for (int kd = 0; kd < 3; ++kd)
    for (int kh = 0; kh < 3; ++kh)
      for (int kw = 0; kw < 3; ++kw) {
        const int id = od * SD - PD + kd;
        const int ih = oh * SH - PH + kh;
        const int iw = ow * SW - PW + kw;
        if (id >= 0 && id < ID && ih >= 0 && ih < IH && iw >= 0 && iw < IW)
          acc += mi[(id * IH + ih) * IW + iw];
      }
  mo[v] = acc > 0.f ? 1.f : 0.f;
}

// x:(2,32,13,128) NDHWC -> out[(w*128+c), d, h]  (reshape+transpose)
__global__ void final_transpose(const float* __restrict__ x,
                                float* __restrict__ out) {
  int i = blockIdx.x * blockDim.x + threadIdx.x;
  const int N = 2 * 32 * 13 * 128;
  if (i >= N) return;
  const int c = i % 128;
  int t = i / 128;
  const int w = t % 13;
  t /= 13;
  const int h = t % 32;
  const int d = t / 32;
  out[((w * 128 + c) * 2 + d) * 32 + h] = x[i];
}

// ---------------------------------------------------------------------------
// Host launcher
// ---------------------------------------------------------------------------
extern "C" void kernel_launch(void* const* d_in, const int* in_sizes, int n_in,
                              void* d_out, int out_size, void* d_ws,
                              size_t ws_size, hipStream_t stream) {
  (void)in_sizes; (void)n_in; (void)out_size; (void)ws_size;
  const int D0 = 11, H0 = 256, W0 = 224;                 // 630784 voxels
  const int D1 = 6, H1 = 128, W1 = 112;                  // 86016
  const int D2 = 3, H2 = 64, W2 = 56;                    // 10752
  const int D3 = 2, H3 = 32, W3 = 27;                    // 1728
  const int D4 = 2, H4 = 32, W4 = 13;                    // 832
  const int NV0 = D0 * H0 * W0, NV1 = D1 * H1 * W1, NV2 = D2 * H2 * W2,
            NV3 = D3 * H3 * W3, NV4 = D4 * H4 * W4;
  const int NPTS = 80000;

  // Inputs flattened in setup_inputs() dict/insertion order.
  auto F = [&](int i) { return (const float*)d_in[i]; };
  const float* features = F(0);
  const int* coors = (const int*)d_in[1];
  // d_in[2] = batch_size (==1)
  const float* w_in = F(3);
  const float *bn_in_s = F(4), *bn_in_b = F(5);
  const int S0B0 = 6, S0B1 = 12, S0S = 18;
  const int S1B0 = 21, S1B1 = 27, S1S = 33;
  const int S2B0 = 36, S2B1 = 42, S2S = 48;
  const int S3B0 = 51, S3B1 = 57;
  const float *bn_out_s = F(64), *bn_out_b = F(65);

  // Workspace bump allocator (256 B aligned)
  char* wsp = (char*)d_ws;
  auto alloc = [&](size_t nfl) {
    float* p = (float*)wsp;
    wsp += ((nfl * sizeof(float) + 255) / 256) * 256;
    return p;
  };
  float* dense4 = alloc((size_t)NV0 * 4);
  float* mask0 = alloc(NV0);
  float* mask1 = alloc(NV1);
  float* mask2 = alloc(NV2);
  float* mask3 = alloc(NV3);
  float* mask4 = alloc(NV4);
  float* win4 = alloc(27 * 4 * 16);
  float* x0a = alloc((size_t)NV0 * 16);
  float* x0b = alloc((size_t)NV0 * 16);
  float* x0c = alloc((size_t)NV0 * 16);
  float* x1a = alloc((size_t)NV1 * 32);
  float* x1b = alloc((size_t)NV1 * 32);
  float* x1c = alloc((size_t)NV1 * 32);
  float* x2a = alloc((size_t)NV2 * 64);
  float* x2b = alloc((size_t)NV2 * 64);
  float* x2c = alloc((size_t)NV2 * 64);
  float* x3a = alloc((size_t)NV3 * 128);
  float* x3b = alloc((size_t)NV3 * 128);
  float* x3c = alloc((size_t)NV3 * 128);
  float* x4 = alloc((size_t)NV4 * 128);

  // Packed weights: {input index, cin, cout}; P[20] in this order.
  struct WSpec { int idx, cin, cout; };
  const WSpec spec[20] = {
      {S0B0 + 0, 16, 16},  {S0B0 + 3, 16, 16},  {S0B1 + 0, 16, 16},
      {S0B1 + 3, 16, 16},  {S0S, 16, 32},
      {S1B0 + 0, 32, 32},  {S1B0 + 3, 32, 32},  {S1B1 + 0, 32, 32},
      {S1B1 + 3, 32, 32},  {S1S, 32, 64},
      {S2B0 + 0, 64, 64},  {S2B0 + 3, 64, 64},  {S2B1 + 0, 64, 64},
      {S2B1 + 3, 64, 64},  {S2S, 64, 128},
      {S3B0 + 0, 128, 128}, {S3B0 + 3, 128, 128}, {S3B1 + 0, 128, 128},
      {S3B1 + 3, 128, 128}, {63, 128, 128}};
  float* P[20];
  float* winP = alloc(27 * 4 * 16);
  for (int i = 0; i < 20; ++i)
    P[i] = alloc((size_t)27 * spec[i].cin * spec[i].cout);

  const dim3 wblk(32);
  auto nb = [&](int n) { return (n + 255) / 256; };
  auto cg = [&](int nvox, int cout, int ntile) {
    return dim3((nvox + 15) / 16, cout / (16 * ntile));
  };

  // ---- weight packing (tiny, runs each call; deterministic) ----
  pad_win<<<nb(27 * 4 * 16), 256, 0, stream>>>(w_in, win4);
  pack_wgt<<<nb(27 * 4 * 16), 256, 0, stream>>>(win4, winP, 4, 16);
  for (int i = 0; i < 20; ++i) {
    const int tot = 27 * spec[i].cin * spec[i].cout;
    pack_wgt<<<nb(tot), 256, 0, stream>>>(F(spec[i].idx), P[i], spec[i].cin,
                                          spec[i].cout);
  }

  // ---- build dense input + mask ----
  zero_f32<<<nb(NV0 * 4), 256, 0, stream>>>(dense4, NV0 * 4);
  zero_f32<<<nb(NV0), 256, 0, stream>>>(mask0, NV0);
  scatter_pts<<<nb(NPTS), 256, 0, stream>>>(features, coors, dense4, mask0,
                                            NPTS, D0, H0, W0);

  // Input conv 4->16, bn+relu+mask
  conv3d_wmma<4, 16, 1><<<cg(NV0, 16, 1), wblk, 0, stream>>>(
      dense4, x0a, winP, bn_in_s, bn_in_b, mask0, nullptr,
      D0, H0, W0, D0, H0, W0, 1, 1, 1, 1, 1, 1, EPI_BN_RELU_MASK);

  // ---- stage 0 (16 ch, full res) ----
  conv3d_wmma<16, 16, 1><<<cg(NV0, 16, 1), wblk, 0, stream>>>(
      x0a, x0b, P[0], F(S0B0 + 1), F(S0B0 + 2), mask0, nullptr,
      D0, H0, W0, D0, H0, W0, 1, 1, 1, 1, 1, 1, EPI_BN_RELU_MASK);
  conv3d_wmma<16, 16, 1><<<cg(NV0, 16, 1), wblk, 0, stream>>>(
      x0b, x0c, P[1], F(S0B0 + 4), F(S0B0 + 5), mask0, x0a,
      D0, H0, W0, D0, H0, W0, 1, 1, 1, 1, 1, 1, EPI_BN_MASK_ADD_RELU);
  conv3d_wmma<16, 16, 1><<<cg(NV0, 16, 1), wblk, 0, stream>>>(
      x0c, x0a, P[2], F(S0B1 + 1), F(S0B1 + 2), mask0, nullptr,
      D0, H0, W0, D0, H0, W0, 1, 1, 1, 1, 1, 1, EPI_BN_RELU_MASK);
  conv3d_wmma<16, 16, 1><<<cg(NV0, 16, 1), wblk, 0, stream>>>(
      x0a, x0b, P[3], F(S0B1 + 4), F(S0B1 + 5), mask0, x0c,
      D0, H0, W0, D0, H0, W0, 1, 1, 1, 1, 1, 1, EPI_BN_MASK_ADD_RELU);
  mask_down<<<nb(NV1), 256, 0, stream>>>(mask0, mask1, D0, H0, W0, D1, H1, W1,
                                         2, 2, 2, 1, 1, 1);
  conv3d_wmma<16, 32, 2><<<cg(NV1, 32, 2), wblk, 0, stream>>>(
      x0b, x1a, P[4], F(S0S + 1), F(S0S + 2), mask1, nullptr,
      D0, H0, W0, D1, H1, W1, 2, 2, 2, 1, 1, 1, EPI_BN_RELU_MASK);

  // ---- stage 1 (32 ch) ----
  conv3d_wmma<32, 32, 2><<<cg(NV1, 32, 2), wblk, 0, stream>>>(
      x1a, x1b, P[5], F(S1B0 + 1), F(S1B0 + 2), mask1, nullptr,
      D1, H1, W1, D1, H1, W1, 1, 1, 1, 1, 1, 1, EPI_BN_RELU_MASK);
  conv3d_wmma<32, 32, 2><<<cg(NV1, 32, 2), wblk, 0, stream>>>(
      x1b, x1c, P[6], F(S1B0 + 4), F(S1B0 + 5), mask1, x1a,
      D1, H1, W1, D1, H1, W1, 1, 1, 1, 1, 1, 1, EPI_BN_MASK_ADD_RELU);
  conv3d_wmma<32, 32, 2><<<cg(NV1, 32, 2), wblk, 0, stream>>>(
      x1c, x1a, P[7], F(S1B1 + 1), F(S1B1 + 2), mask1, nullptr,
      D1, H1, W1, D1, H1, W1, 1, 1, 1, 1, 1, 1, EPI_BN_RELU_MASK);
  conv3d_wmma<32, 32, 2><<<cg(NV1, 32, 2), wblk, 0, stream>>>(
      x1a, x1b, P[8], F(S1B1 + 4), F(S1B1 + 5), mask1, x1c,
      D1, H1, W1, D1, H1, W1, 1, 1, 1, 1, 1, 1, EPI_BN_MASK_ADD_RELU);
  mask_down<<<nb(NV2), 256, 0, stream>>>(mask1, mask2, D1, H1, W1, D2, H2, W2,
                                         2, 2, 2, 1, 1, 1);
  conv3d_wmma<32, 64, 4><<<cg(NV2, 64, 4), wblk, 0, stream>>>(
      x1b, x2a, P[9], F(S1S + 1), F(S1S + 2), mask2, nullptr,
      D1, H1, W1, D2, H2, W2, 2, 2, 2, 1, 1, 1, EPI_BN_RELU_MASK);

  // ---- stage 2 (64 ch) ----
  conv3d_wmma<64, 64, 4><<<cg(NV2, 64, 4), wblk, 0, stream>>>(
      x2a, x2b, P[10], F(S2B0 + 1), F(S2B0 + 2), mask2, nullptr,
      D2, H2, W2, D2, H2, W2, 1, 1, 1, 1, 1, 1, EPI_BN_RELU_MASK);
  conv3d_wmma<64, 64, 4><<<cg(NV2, 64, 4), wblk, 0, stream>>>(
      x2b, x2c, P[11], F(S2B0 + 4), F(S2B0 + 5), mask2, x2a,
      D2, H2, W2, D2, H2, W2, 1, 1, 1, 1, 1, 1, EPI_BN_MASK_ADD_RELU);
  conv3d_wmma<64, 64, 4><<<cg(NV2, 64, 4), wblk, 0, stream>>>(
      x2c, x2a, P[12], F(S2B1 + 1), F(S2B1 + 2), mask2, nullptr,
      D2, H2, W2, D2, H2, W2, 1, 1, 1, 1, 1, 1, EPI_BN_RELU_MASK);
  conv3d_wmma<64, 64, 4><<<cg(NV2, 64, 4), wblk, 0, stream>>>(
      x2a, x2b, P[13], F(S2B1 + 4), F(S2B1 + 5), mask2, x2c,
      D2, H2, W2, D2, H2, W2, 1, 1, 1, 1, 1, 1, EPI_BN_MASK_ADD_RELU);
  mask_down<<<nb(NV3), 256, 0, stream>>>(mask2, mask3, D2, H2, W2, D3, H3, W3,
                                         2, 2, 2, 1, 1, 0);
  conv3d_wmma<64, 128, 4><<<cg(NV3, 128, 4), wblk, 0, stream>>>(
      x2b, x3a, P[14], F(S2S + 1), F(S2S + 2), mask3, nullptr,
      D2, H2, W2, D3, H3, W3, 2, 2, 2, 1, 1, 0, EPI_BN_RELU_MASK);

  // ---- stage 3 (128 ch) ----
  conv3d_wmma<128, 128, 4><<<cg(NV3, 128, 4), wblk, 0, stream>>>(
      x3a, x3b, P[15], F(S3B0 + 1), F(S3B0 + 2), mask3, nullptr,
      D3, H3, W3, D3, H3, W3, 1, 1, 1, 1, 1, 1, EPI_BN_RELU_MASK);
  conv3d_wmma<128, 128, 4><<<cg(NV3, 128, 4), wblk, 0, stream>>>(
      x3b, x3c, P[16], F(S3B0 + 4), F(S3B0 + 5), mask3, x3a,
      D3, H3, W3, D3, H3, W3, 1, 1, 1, 1, 1, 1, EPI_BN_MASK_ADD_RELU);
  conv3d_wmma<128, 128, 4><<<cg(NV3, 128, 4), wblk, 0, stream>>>(
      x3c, x3a, P[17], F(S3B1 + 1), F(S3B1 + 2), mask3, nullptr,
      D3, H3, W3, D3, H3, W3, 1, 1, 1, 1, 1, 1, EPI_BN_RELU_MASK);
  conv3d_wmma<128, 128, 4><<<cg(NV3, 128, 4), wblk, 0, stream>>>(
      x3a, x3b, P[18], F(S3B1 + 4), F(S3B1 + 5), mask3, x3c,
      D3, H3, W3, D3, H3, W3, 1, 1, 1, 1, 1, 1, EPI_BN_MASK_ADD_RELU);

  // ---- output conv, stride (1,1,2), pad (1,1,0) ----
  mask_down<<<nb(NV4), 256, 0, stream>>>(mask3, mask4, D3, H3, W3, D4, H4, W4,
                                         1, 1, 2, 1, 1, 0);
  conv3d_wmma<128, 128, 4><<<cg(NV4, 128, 4), wblk, 0, stream>>>(
      x3b, x4, P[19], bn_out_s, bn_out_b, mask4, nullptr,
      D3, H3, W3, D4, H4, W4, 1, 1, 2, 1, 1, 0, EPI_BN_RELU_MASK);

  final_transpose<<<nb(NV4 * 128), 256, 0, stream>>>(x4, (float*)d_out);
}